// EnergyBorn_33543694582097
// MI455X (gfx1250) — compile-verified
//
#include <hip/hip_runtime.h>
#include <cstdint>

#define N_MOL_C   128
#define Z_MAX_C   90
#define TABLE_ELEMS (2 * 2 * Z_MAX_C * Z_MAX_C)   // 32400 floats = 129,600 B
#define CUTOFF2   25.0f
#define HALF_KE   7.1998f                          // 0.5 * 14.3996

__device__ __forceinline__ float fast_exp2(float x) {
#if __has_builtin(__builtin_amdgcn_exp2f)
    return __builtin_amdgcn_exp2f(x);              // v_exp_f32
#else
    return exp2f(x);
#endif
}

__device__ __forceinline__ float fast_log2(float x) {
#if __has_builtin(__builtin_amdgcn_logf)
    return __builtin_amdgcn_logf(x);               // v_log_f32 (base-2)
#else
    return __log2f(x);
#endif
}

struct __align__(16) AtomPack {
    float q;     // partial charge
    float ns;    // Born exponent
    int   a_m;   // (idx_m << 16) | (is_film*2*90*90 + Z*90)   (low part <= 24210)
    int   b;     // is_film*90*90 + Z                           (<= 8189)
};

struct F3 { float x, y, z; };                      // 12B, align 4 -> b96 load

// Prologue: pack per-atom data into one 16B record; also zero the 128 outputs
// (harness poisons d_out, and we accumulate with atomics).
__global__ void born_pack_atoms(const float* __restrict__ q,
                                const int*   __restrict__ Z,
                                const float* __restrict__ ns,
                                const int*   __restrict__ idx_m,
                                const int*   __restrict__ is_film,
                                AtomPack*    __restrict__ packed,
                                float*       __restrict__ out,
                                int nAtoms) {
    int t = blockIdx.x * blockDim.x + threadIdx.x;
    if (t < N_MOL_C) out[t] = 0.0f;
    if (t >= nAtoms) return;
    AtomPack p;
    p.q   = q[t];
    p.ns  = ns[t];
    int f = is_film[t];
    int z = Z[t];
    p.a_m = (idx_m[t] << 16) | (f * (2 * Z_MAX_C * Z_MAX_C) + z * Z_MAX_C);
    p.b   = f * (Z_MAX_C * Z_MAX_C) + z;
    packed[t] = p;
}

__global__ __launch_bounds__(256) void born_edges(
    const AtomPack* __restrict__ packed,
    const F3*       __restrict__ rij,      // [E] of 12B
    const int*      __restrict__ idx_i,
    const int*      __restrict__ idx_j,
    const float*    __restrict__ r0_table, // 2*2*90*90
    float*          __restrict__ out,      // [128]
    int nEdges) {
    __shared__ __align__(16) float tab[TABLE_ELEMS];
    __shared__ float acc[N_MOL_C];

    for (int m = threadIdx.x; m < N_MOL_C; m += blockDim.x) acc[m] = 0.0f;

    // ---- CDNA5 async copy: r0 table (global -> LDS), 16B per lane per op ----
    {
        const int nVec = TABLE_ELEMS / 4;          // 8100 float4
        const float4* src = (const float4*)r0_table;
        for (int k = threadIdx.x; k < nVec; k += blockDim.x) {
            unsigned ldsOff = (unsigned)(uintptr_t)(&tab[k * 4]);
            unsigned long long ga = (unsigned long long)(uintptr_t)(src + k);
            asm volatile("global_load_async_to_lds_b128 %0, %1, off"
                         :: "v"(ldsOff), "v"(ga) : "memory");
        }
        asm volatile("s_wait_asynccnt 0x0" ::: "memory");
    }
    __syncthreads();

    const float L2_CUTOFF = 2.321928094887362f;    // log2(5)
    const int stride = gridDim.x * blockDim.x;
#pragma unroll 4
    for (int e = blockIdx.x * blockDim.x + threadIdx.x; e < nEdges; e += stride) {
        int i = idx_i[e];
        int j = idx_j[e];
        AtomPack Ai = packed[i];                   // global_load_b128 (L2-resident)
        AtomPack Aj = packed[j];
        F3 R = rij[e];                             // global_load_b96, coalesced
        float d2 = R.x * R.x + R.y * R.y + R.z * R.z;

        float n   = Ai.ns + 0.5f * Aj.ns;          // ns_i + ns_j/2
        int   mol = Ai.a_m >> 16;
        float r0  = tab[(Ai.a_m & 0xFFFF) + Aj.b]; // ds_load_b32
        float qij = fabsf(Ai.q * Aj.q);

        // d^-n = exp2(-n * 0.5*log2(d^2))  (no sqrt needed)
        float l2d  = 0.5f * fast_log2(d2);
        float B    = qij * fast_exp2((n - 1.0f) * fast_log2(r0)) *
                     __builtin_amdgcn_rcpf(n);
        float pot  = B * (fast_exp2(-n * l2d) - fast_exp2(-n * L2_CUTOFF));

        if (d2 <= CUTOFF2)
            atomicAdd(&acc[mol], pot);             // ds_add_f32
    }
    __syncthreads();

    for (int m = threadIdx.x; m < N_MOL_C; m += blockDim.x)
        atomicAdd(&out[m], HALF_KE * acc[m]);      // global_atomic_add_f32
}

extern "C" void kernel_launch(void* const* d_in, const int* in_sizes, int n_in,
                              void* d_out, int out_size, void* d_ws, size_t ws_size,
                              hipStream_t stream) {
    const float* q       = (const float*)d_in[0];
    const int*   Z       = (const int*)  d_in[1];
    const float* ns      = (const float*)d_in[2];
    const int*   idx_m   = (const int*)  d_in[3];
    const F3*    rij     = (const F3*)   d_in[4];
    const int*   idx_i   = (const int*)  d_in[5];
    const int*   idx_j   = (const int*)  d_in[6];
    const int*   is_film = (const int*)  d_in[7];
    const float* r0_tab  = (const float*)d_in[8];
    float* out = (float*)d_out;

    int nAtoms = in_sizes[0];
    int nEdges = in_sizes[5];

    AtomPack* packed = (AtomPack*)d_ws;            // 100K * 16B = 1.6 MB scratch

    int packBlocks = (nAtoms + 255) / 256;
    born_pack_atoms<<<packBlocks, 256, 0, stream>>>(q, Z, ns, idx_m, is_film,
                                                    packed, out, nAtoms);
    born_edges<<<1280, 256, 0, stream>>>(packed, rij, idx_i, idx_j, r0_tab,
                                         out, nEdges);
}